// TripletLoss_1477468750224
// MI455X (gfx1250) — compile-verified
//
#include <hip/hip_runtime.h>
#include <math.h>

typedef __attribute__((ext_vector_type(2))) float v2f;
typedef __attribute__((ext_vector_type(4))) float v4f;
typedef __attribute__((ext_vector_type(8))) float v8f;

#define NB    62          // batches (parts)
#define MM    256         // rows per batch
#define DIM   256         // feature dim
#define KP    8           // positives per anchor (incl. self)
#define NNEG  (MM - KP)   // 248
#define MARGIN 0.2f
#define NTILE 136         // 16*17/2 triangular 16x16 tiles (ti <= tj)

// Output layout (floats, total 30815):
//   [0, NB*NNEG)                       full_mean   (62,248)
//   [NB*NNEG, NB*NNEG+NB)              hard mean   (62,)
//   [NB*NNEG+NB]                       mean_dist   scalar
//   [NB*NNEG+NB+1, ... +NB*NNEG)       full_num    (62,248)

// One workgroup per batch n. 256 threads = 8 wave32s.
// Dynamic LDS layout (floats then ints):
//   Dl[65536] | x2[256] | posd[256*8] | redA[256] | redB[256] | lab[256] | pst[256]
__global__ __launch_bounds__(256) void triplet_main(const float* __restrict__ feature,
                                                    const int*   __restrict__ label,
                                                    float*       __restrict__ out,
                                                    float*       __restrict__ ws)
{
    extern __shared__ float smem[];
    float* Dl   = smem;                 // 65536 : full 256x256 distance matrix
    float* x2   = Dl + MM * MM;         // 256   : row squared norms
    float* posd = x2 + MM;              // 2048  : per-row positive distances
    float* redA = posd + MM * KP;       // 256   : reduction scratch (hard loss)
    float* redB = redA + MM;            // 256   : reduction scratch (dist sum)
    int*   lab  = (int*)(redB + MM);    // 256   : labels
    int*   pst  = (int*)(lab + MM);     // 256   : first positive column per row

    const int n   = blockIdx.x;
    const int tid = threadIdx.x;
    const float* F = feature + (size_t)n * MM * DIM;

    // ---- phase 1: row norms + labels -> LDS ----
    {
        const v4f* row = (const v4f*)(F + tid * DIM);
        float s = 0.f;
#pragma unroll 4
        for (int q = 0; q < DIM / 4; ++q) {
            v4f v = row[q];
            s += v.x * v.x + v.y * v.y + v.z * v.z + v.w * v.w;
        }
        x2[tid]  = s;
        lab[tid] = label[n * MM + tid];
    }
    __syncthreads();

    // ---- phase 2: WMMA f32 Gram (triangular, mirrored) -> distances in LDS ----
    // A 16x4 frag: lane L -> row L%16, K = 2*(L/16)+{0,1} (one contiguous f32x2)
    // B 4x16 frag (B = F^T): identical addressing on the column tile's rows.
    // 136 = 8 waves * 17 tiles: uniform trip count, EXEC all-ones through WMMA.
    {
        const int wave = tid >> 5;
        const int lane = tid & 31;
        const int lm   = lane & 15;   // row within tile
        const int lh   = lane >> 4;   // K half select
        for (int t = wave; t < NTILE; t += 8) {
            // decode triangular index -> (ti, tj), ti <= tj (wave-uniform)
            int rem = t, ti = 0;
            while (rem >= 16 - ti) { rem -= 16 - ti; ++ti; }
            const int tj = ti + rem;

            const float* Arow = F + (ti * 16 + lm) * DIM + 2 * lh;
            const float* Brow = F + (tj * 16 + lm) * DIM + 2 * lh;
            v8f c = {};
#pragma unroll 16
            for (int k = 0; k < DIM; k += 4) {
                v2f a = *(const v2f*)(Arow + k);
                v2f b = *(const v2f*)(Brow + k);
                // 8 args: (neg_a, A, neg_b, B, c_mod, C, reuse_a, reuse_b)
                c = __builtin_amdgcn_wmma_f32_16x16x4_f32(
                        false, a, false, b, (short)0, c, false, false);
            }
            // C/D layout: VGPR r, lane L -> M = r + 8*(L/16), N = L%16
#pragma unroll
            for (int r = 0; r < 8; ++r) {
                int mi = ti * 16 + r + 8 * lh;
                int nj = tj * 16 + lm;
                float d2 = x2[mi] + x2[nj] - 2.0f * c[r];
                d2 = fmaxf(d2, 0.0f) + 1e-12f;
                float dv = sqrtf(d2);
                Dl[mi * MM + nj] = dv;
                Dl[nj * MM + mi] = dv;   // symmetric mirror (bitwise-identical value)
            }
        }
    }
    __syncthreads();

    // ---- phase 3: per-row hard mining + positive extraction (1 thread = 1 row) ----
    {
        const int i = tid;
        const int g = lab[i];
        float pmax = -1e30f, nmin = 1e30f, rs = 0.f;
        int pc = 0, ps = MM;
        const float* Drow = Dl + i * MM;
        for (int j = 0; j < MM; ++j) {
            float dv = Drow[j];
            rs += dv;
            if (lab[j] == g) {          // stable column-order extraction
                if (pc == 0) ps = j;
                posd[i * KP + pc] = dv;
                ++pc;
                pmax = fmaxf(pmax, dv);
            } else {
                nmin = fminf(nmin, dv);
            }
        }
        pst[i]  = ps;
        redA[i] = fmaxf(MARGIN + pmax - nmin, 0.0f);
        redB[i] = rs;
    }
    __syncthreads();
    // tree reductions: hard-loss sum and distance sum
    for (int off = MM / 2; off > 0; off >>= 1) {
        if (tid < off) {
            redA[tid] += redA[tid + off];
            redB[tid] += redB[tid + off];
        }
        __syncthreads();
    }
    if (tid == 0) {
        out[NB * NNEG + n] = redA[0] * (1.0f / MM);  // hard_loss_metric_mean[n]
        ws[n] = redB[0];                             // partial distance sum
    }

    // ---- phase 4: full mining. thread k accumulates over all (i, p) ----
    if (tid < NNEG) {
        const int k = tid;
        float sum = 0.f, cnt = 0.f;
        for (int i = 0; i < MM; ++i) {
            // k-th negative column: skip the contiguous positive block of this row
            int j = (k < pst[i]) ? k : (k + KP);
            float base = MARGIN - Dl[i * MM + j];
            const float* pp = posd + i * KP;
#pragma unroll
            for (int p = 0; p < KP; ++p) {
                float v = base + pp[p];
                if (v > 0.f) { sum += v; cnt += 1.f; }
            }
        }
        out[n * NNEG + k]                      = sum / (cnt + 1e-6f);  // full_mean
        out[NB * NNEG + NB + 1 + n * NNEG + k] = cnt;                  // full_num
    }
}

// Deterministic fixed-order reduction of per-batch distance sums -> mean_dist.
__global__ void triplet_finish(const float* __restrict__ ws, float* __restrict__ out)
{
    if (threadIdx.x == 0 && blockIdx.x == 0) {
        float s = 0.f;
        for (int i = 0; i < NB; ++i) s += ws[i];
        out[NB * NNEG + NB] = s * (1.0f / ((float)NB * (float)MM * (float)MM));
    }
}

extern "C" void kernel_launch(void* const* d_in, const int* in_sizes, int n_in,
                              void* d_out, int out_size, void* d_ws, size_t ws_size,
                              hipStream_t stream)
{
    const float* feature = (const float*)d_in[0];
    const int*   label   = (const int*)d_in[1];
    // d_in[2] = num_pos (always 8 for this problem; compiled in as KP)
    float* out = (float*)d_out;
    float* ws  = (float*)d_ws;   // uses NB floats only

    const size_t lds_bytes =
        (size_t)(MM * MM + MM + MM * KP + MM + MM) * sizeof(float) +
        (size_t)(2 * MM) * sizeof(int);   // ~269 KB, fits 320 KB WGP LDS

    triplet_main<<<NB, 256, lds_bytes, stream>>>(feature, label, out, ws);
    triplet_finish<<<1, 32, 0, stream>>>(ws, out);
}